// MLoss_76699525971982
// MI455X (gfx1250) — compile-verified
//
#include <hip/hip_runtime.h>
#include <math.h>

typedef float v2f __attribute__((ext_vector_type(2)));
typedef float v8f __attribute__((ext_vector_type(8)));

#define NB      1024      // pass-1 blocks; must be a multiple of 64 for the WMMA pass
#define THREADS 256

// Per-cell loss terms, branchless. Accumulates into cnt/sq/bce/bg.
#define CELL(P, T, X0, X1, X2, X3, Y0, Y1, Y2, Y3)                            \
  do {                                                                        \
    float m_  = ((T) > 0.5f) ? 1.0f : 0.0f;                                   \
    float e0_ = (X0) - (Y0), e1_ = (X1) - (Y1);                               \
    float e2_ = (X2) - (Y2), e3_ = (X3) - (Y3);                               \
    float s_  = e0_ * e0_ + e1_ * e1_ + e2_ * e2_ + e3_ * e3_;                \
    float lp_ = fmaxf(logf((P)), -100.0f);                                    \
    float l1_ = fmaxf(logf(1.0f - (P)), -100.0f);                             \
    float b_  = -((T) * lp_ + (1.0f - (T)) * l1_);                            \
    cnt += m_;                                                                \
    sq  += m_ * s_;                                                           \
    bce += m_ * b_;                                                           \
    bg  += (1.0f - m_) * (-l1_);                                              \
  } while (0)

// Pass 1: stream x,y once (b128 loads), per-block partial (cnt, sq, bce, bg).
// Partials stored transposed: ws[0..NB) = cnt, ws[NB..2NB) = sq, etc.
__global__ __launch_bounds__(THREADS) void mloss_partial(
    const float* __restrict__ x, const float* __restrict__ y,
    float* __restrict__ ws, int cells) {
  const int tid    = threadIdx.x;
  const int gtid   = blockIdx.x * blockDim.x + tid;
  const int stride = gridDim.x * blockDim.x;
  const int groups = cells >> 2;  // 4 cells = 20 floats = 5 x float4 (16B aligned)

  float cnt = 0.0f, sq = 0.0f, bce = 0.0f, bg = 0.0f;

  for (int g = gtid; g < groups; g += stride) {
    // Prefetch next grid-stride iteration (emits global_prefetch_b8).
    const int gn = g + stride;
    if (gn < groups) {
      __builtin_prefetch(x + 20 * gn, 0, 0);
      __builtin_prefetch(y + 20 * gn, 0, 0);
    }
    const float4* xp = (const float4*)(x + 20 * g);
    const float4* yp = (const float4*)(y + 20 * g);
    float4 a0 = xp[0], a1 = xp[1], a2 = xp[2], a3 = xp[3], a4 = xp[4];
    float4 b0 = yp[0], b1 = yp[1], b2 = yp[2], b3 = yp[3], b4 = yp[4];

    CELL(a0.x, b0.x, a0.y, a0.z, a0.w, a1.x, b0.y, b0.z, b0.w, b1.x);
    CELL(a1.y, b1.y, a1.z, a1.w, a2.x, a2.y, b1.z, b1.w, b2.x, b2.y);
    CELL(a2.z, b2.z, a2.w, a3.x, a3.y, a3.z, b2.w, b3.x, b3.y, b3.z);
    CELL(a3.w, b3.w, a4.x, a4.y, a4.z, a4.w, b3.w == b3.w ? b4.x : b4.x, b4.y, b4.z, b4.w);
  }

  // Scalar tail for cells % 4 (zero for the reference shape).
  for (int c = (groups << 2) + gtid; c < cells; c += stride) {
    const float* xc = x + 5 * c;
    const float* yc = y + 5 * c;
    CELL(xc[0], yc[0], xc[1], xc[2], xc[3], xc[4], yc[1], yc[2], yc[3], yc[4]);
  }

  // Block tree-reduction (deterministic).
  __shared__ float4 red[THREADS];
  red[tid] = make_float4(cnt, sq, bce, bg);
  __syncthreads();
  for (int off = THREADS >> 1; off > 0; off >>= 1) {
    if (tid < off) {
      float4 o = red[tid + off];
      float4 m = red[tid];
      m.x += o.x; m.y += o.y; m.z += o.z; m.w += o.w;
      red[tid] = m;
    }
    __syncthreads();
  }
  if (tid == 0) {
    float4 r = red[0];
    ws[blockIdx.x + 0 * NB] = r.x;
    ws[blockIdx.x + 1 * NB] = r.y;
    ws[blockIdx.x + 2 * NB] = r.z;
    ws[blockIdx.x + 3 * NB] = r.w;
  }
}

// Pass 2: 4 waves, wave w reduces partial array w with V_WMMA_F32_16X16X4_F32.
// B = ones(4x16)  =>  D[i][j] += sum_k A[i][k], i.e. each WMMA folds a 64-float
// chunk (A 16x4 layout: lane l<16 holds A[l][0..1], lane l+16 holds A[l][2..3]).
__global__ __launch_bounds__(128) void mloss_final(
    const float* __restrict__ ws, float* __restrict__ out, int cells) {
  __shared__ float totals[4];
  const int lane = threadIdx.x & 31;
  const int wave = threadIdx.x >> 5;
  const float* arr = ws + wave * NB;

  v2f ones = {1.0f, 1.0f};
  v8f d = {0.0f, 0.0f, 0.0f, 0.0f, 0.0f, 0.0f, 0.0f, 0.0f};
  const int off0 = 4 * (lane & 15) + 2 * (lane >> 4);

  for (int ch = 0; ch < NB; ch += 64) {
    float2 v = *(const float2*)(arr + ch + off0);
    v2f a;
    a[0] = v.x;
    a[1] = v.y;
    // D = A(16x4) x B(4x16 ones) + C   -> row sums accumulate in every column
    d = __builtin_amdgcn_wmma_f32_16x16x4_f32(
        /*neg_a=*/false, a, /*neg_b=*/false, ones,
        /*c_mod=*/(short)0, d, /*reuse_a=*/false, /*reuse_b=*/false);
  }

  // Column 0 of D: rows 0..7 live in lane 0, rows 8..15 in lane 16.
  float s = d[0] + d[1] + d[2] + d[3] + d[4] + d[5] + d[6] + d[7];
  float tot = __shfl(s, 0, 32) + __shfl(s, 16, 32);
  if (lane == 0) totals[wave] = tot;
  __syncthreads();

  if (threadIdx.x == 0) {
    float face  = totals[0];
    float sqs   = totals[1];
    float bces  = totals[2];
    float bgs   = totals[3];
    float bgn   = (float)cells - face;
    float scale = 1.0f + 1.0f / face;
    float diff_box = scale * sqs / (face * 4.0f);
    float diff_c   = scale * bces / face;
    float diff_bg  = bgs / bgn;  // ALPHA = 1
    out[0] = diff_box + diff_c + diff_bg;
  }
}

extern "C" void kernel_launch(void* const* d_in, const int* in_sizes, int n_in,
                              void* d_out, int out_size, void* d_ws, size_t ws_size,
                              hipStream_t stream) {
  const float* x = (const float*)d_in[0];
  const float* y = (const float*)d_in[1];
  float* out = (float*)d_out;
  float* ws  = (float*)d_ws;  // needs 4 * NB * 4 = 16 KB
  const int cells = in_sizes[0] / 5;  // [B, N, 5] -> B*N

  mloss_partial<<<NB, THREADS, 0, stream>>>(x, y, ws, cells);
  mloss_final<<<1, 128, 0, stream>>>(ws, out, cells);
}